// InsideLoss2D_86517821214300
// MI455X (gfx1250) — compile-verified
//
#include <hip/hip_runtime.h>
#include <hip/hip_bf16.h>
#include <float.h>

// CDNA5 / gfx1250: wave32, WMMA f32 16x16x4 for the K=3 fp32 distance GEMM.
// D = |q|^2 - 2 p.q  (C seeded with |q|^2, A scaled by -2): argmin-equivalent
// to squared L2 distance, computed 16x16 per WMMA.

typedef __attribute__((ext_vector_type(2))) float v2f;
typedef __attribute__((ext_vector_type(8))) float v8f;

#define BATCHES 4
#define MCAGE 128
#define ITP 10
#define PQ (MCAGE * ITP)            // 1280 query points per batch
#define ROWS_TOTAL (BATCHES * PQ)   // 5120
#define NPTS 20000
#define EPSI 0.01f
#define NT 256                      // shape points staged in LDS per block iteration
#define WAVES 8                     // waves per block; each wave owns one 16-row M-tile
#define BLOCKS_PER_BATCH (PQ / 16 / WAVES)  // 10
#define NSPLIT 8
#define NPER (NPTS / NSPLIT)        // 2500

// ---------------- common device helpers ----------------

__device__ __forceinline__ void query_point(const float* __restrict__ cage,
                                            int batch, int pIdx,
                                            float& px, float& py, float& pz) {
    const int edge = pIdx / ITP;
    const int kk   = pIdx % ITP;
    const float t  = (float)kk * (1.0f / (float)(ITP - 1));
    const float* c0 = cage + ((size_t)batch * MCAGE + edge) * 3;
    const float* c1 = cage + ((size_t)batch * MCAGE + ((edge + 1) & (MCAGE - 1))) * 3;
    px = t * c1[0] + (1.0f - t) * c0[0];
    py = t * c1[1] + (1.0f - t) * c0[1];
    pz = t * c1[2] + (1.0f - t) * c0[2];
}

// Scan shape points [nStart, nEnd) with 16x16x4 f32 WMMA tiles; per-lane
// running (min, argmin) over this lane's N-residue class, per M-row VGPR.
__device__ __forceinline__ void wmma_argmin_scan(
    const float* __restrict__ shapeB, int nStart, int nEnd,
    v2f a, int lane,
    float* sX, float* sY, float* sZ, float* sN,
    float dmin[8], int imin[8])
{
    for (int nbase = nStart; nbase < nEnd; nbase += NT) {
        const int n = nbase + (int)threadIdx.x;
        float qx = 0.0f, qy = 0.0f, qz = 0.0f, qn = 3.0e38f;  // pad never wins
        if (n < nEnd) {
            qx = shapeB[n * 3 + 0];
            qy = shapeB[n * 3 + 1];
            qz = shapeB[n * 3 + 2];
            qn = qx * qx + qy * qy + qz * qz;
            if (n + NT < nEnd) __builtin_prefetch(&shapeB[(n + NT) * 3], 0, 0);
        }
        sX[threadIdx.x] = qx; sY[threadIdx.x] = qy;
        sZ[threadIdx.x] = qz; sN[threadIdx.x] = qn;
        __syncthreads();

#pragma unroll
        for (int sub = 0; sub < NT / 16; ++sub) {
            const int nl = sub * 16 + (lane & 15);
            // B (4x16): lanes 0-15 -> (qx,qy), lanes 16-31 -> (qz,0), col N = lane&15
            v2f b;
            b.x = (lane < 16) ? sX[nl] : sZ[nl];
            b.y = (lane < 16) ? sY[nl] : 0.0f;
            const float cv = sN[nl];
            v8f c = {cv, cv, cv, cv, cv, cv, cv, cv};
            v8f d = __builtin_amdgcn_wmma_f32_16x16x4_f32(
                false, a, false, b, (short)0, c, false, false);
            const int nAbs = nbase + sub * 16 + (lane & 15);
#pragma unroll
            for (int v = 0; v < 8; ++v) {
                const float dv = d[v];
                if (dv < dmin[v]) { dmin[v] = dv; imin[v] = nAbs; }  // ascending: first-min kept
            }
        }
        __syncthreads();
    }
}

// Argmin across the 16 lanes of each half-wave, then hand row r's (d, idx)
// to lane r (which holds the query point for row r = lane & 15).
__device__ __forceinline__ void reduce_rows(float dmin[8], int imin[8], int lane,
                                            float& myD, int& myIdx)
{
#pragma unroll
    for (int off = 1; off < 16; off <<= 1) {
#pragma unroll
        for (int v = 0; v < 8; ++v) {
            const float dO = __shfl_xor(dmin[v], off, 32);
            const int   iO = __shfl_xor(imin[v], off, 32);
            if (dO < dmin[v] || (dO == dmin[v] && iO < imin[v])) {
                dmin[v] = dO; imin[v] = iO;
            }
        }
    }
    myD = FLT_MAX; myIdx = 0;
#pragma unroll
    for (int r = 0; r < 16; ++r) {
        const int src = (r < 8) ? 0 : 16;
        const float dr = __shfl(dmin[r & 7], src, 32);
        const int   ir = __shfl(imin[r & 7], src, 32);
        if (lane == r) { myD = dr; myIdx = ir; }
    }
}

// ---------------- Path A: 8-way N-split (needs ~320KB workspace) ----------------

__global__ __launch_bounds__(256) void split_scan_kernel(
    const float* __restrict__ cage, const float* __restrict__ shape,
    float* __restrict__ dpart, int* __restrict__ ipart)
{
    __shared__ float sX[NT], sY[NT], sZ[NT], sN[NT];
    const int split = blockIdx.x % NSPLIT;
    const int rem   = blockIdx.x / NSPLIT;
    const int batch = rem / BLOCKS_PER_BATCH;
    const int blk   = rem % BLOCKS_PER_BATCH;
    const int wave  = threadIdx.x >> 5;
    const int lane  = threadIdx.x & 31;
    const int rowBase = (blk * WAVES + wave) * 16;

    float px, py, pz;
    query_point(cage, batch, rowBase + (lane & 15), px, py, pz);
    v2f a;
    a.x = (lane < 16) ? (-2.0f * px) : (-2.0f * pz);
    a.y = (lane < 16) ? (-2.0f * py) : 0.0f;

    float dmin[8]; int imin[8];
#pragma unroll
    for (int v = 0; v < 8; ++v) { dmin[v] = FLT_MAX; imin[v] = 0; }

    const float* shapeB = shape + (size_t)batch * NPTS * 3;
    wmma_argmin_scan(shapeB, split * NPER, (split + 1) * NPER, a, lane,
                     sX, sY, sZ, sN, dmin, imin);

    float myD; int myIdx;
    reduce_rows(dmin, imin, lane, myD, myIdx);
    if (lane < 16) {
        const int gRow = batch * PQ + rowBase + lane;
        dpart[gRow * NSPLIT + split] = myD;
        ipart[gRow * NSPLIT + split] = myIdx;
    }
}

__global__ __launch_bounds__(256) void merge_kernel(
    const float* __restrict__ cage, const float* __restrict__ shape,
    const float* __restrict__ normals,
    const float* __restrict__ dpart, const int* __restrict__ ipart,
    float* __restrict__ blockSums)
{
    __shared__ float wsum[WAVES];
    const int g = blockIdx.x * 256 + (int)threadIdx.x;   // < 5120 exactly
    const int batch = g / PQ;
    const int p     = g % PQ;

    float px, py, pz;
    query_point(cage, batch, p, px, py, pz);

    float best = FLT_MAX; int bidx = 0;
#pragma unroll
    for (int s = 0; s < NSPLIT; ++s) {   // ascending split order: first-min kept
        const float d = dpart[g * NSPLIT + s];
        const int   i = ipart[g * NSPLIT + s];
        if (d < best) { best = d; bidx = i; }
    }

    const float* sp = shape   + ((size_t)batch * NPTS + bidx) * 3;
    const float* np = normals + ((size_t)batch * NPTS + bidx) * 3;
    const float nx = np[0], ny = np[1], nz = np[2];
    const float dot = (px - sp[0] - EPSI * nx) * nx
                    + (py - sp[1] - EPSI * ny) * ny
                    + (pz - sp[2] - EPSI * nz) * nz;
    float loss = (dot < 0.0f) ? -dot : 0.0f;

#pragma unroll
    for (int off = 16; off >= 1; off >>= 1) loss += __shfl_xor(loss, off, 32);
    const int lane = threadIdx.x & 31, wave = threadIdx.x >> 5;
    if (lane == 0) wsum[wave] = loss;
    __syncthreads();
    if (threadIdx.x == 0) {
        float s = 0.0f;
        for (int i = 0; i < WAVES; ++i) s += wsum[i];
        blockSums[blockIdx.x] = s;
    }
}

// ---------------- Path B: single-pass fallback (tiny workspace) ----------------

__global__ __launch_bounds__(256) void inside_loss_kernel(
    const float* __restrict__ cage, const float* __restrict__ shape,
    const float* __restrict__ normals, float* __restrict__ blockSums)
{
    __shared__ float sX[NT], sY[NT], sZ[NT], sN[NT];
    __shared__ float wsum[WAVES];
    const int batch = blockIdx.x / BLOCKS_PER_BATCH;
    const int blk   = blockIdx.x % BLOCKS_PER_BATCH;
    const int wave  = threadIdx.x >> 5;
    const int lane  = threadIdx.x & 31;
    const int rowBase = (blk * WAVES + wave) * 16;

    float px, py, pz;
    query_point(cage, batch, rowBase + (lane & 15), px, py, pz);
    v2f a;
    a.x = (lane < 16) ? (-2.0f * px) : (-2.0f * pz);
    a.y = (lane < 16) ? (-2.0f * py) : 0.0f;

    float dmin[8]; int imin[8];
#pragma unroll
    for (int v = 0; v < 8; ++v) { dmin[v] = FLT_MAX; imin[v] = 0; }

    const float* shapeB = shape + (size_t)batch * NPTS * 3;
    wmma_argmin_scan(shapeB, 0, NPTS, a, lane, sX, sY, sZ, sN, dmin, imin);

    float myD; int myIdx;
    reduce_rows(dmin, imin, lane, myD, myIdx);

    float loss = 0.0f;
    if (lane < 16) {
        const float* sp = shapeB + (size_t)myIdx * 3;
        const float* np = normals + ((size_t)batch * NPTS + myIdx) * 3;
        const float nx = np[0], ny = np[1], nz = np[2];
        const float dot = (px - sp[0] - EPSI * nx) * nx
                        + (py - sp[1] - EPSI * ny) * ny
                        + (pz - sp[2] - EPSI * nz) * nz;
        loss = (dot < 0.0f) ? -dot : 0.0f;
    }
#pragma unroll
    for (int off = 16; off >= 1; off >>= 1) loss += __shfl_xor(loss, off, 32);
    if (lane == 0) wsum[wave] = loss;
    __syncthreads();
    if (threadIdx.x == 0) {
        float s = 0.0f;
        for (int i = 0; i < WAVES; ++i) s += wsum[i];
        blockSums[blockIdx.x] = s;
    }
}

// ---------------- deterministic final reduction ----------------

__global__ void finalize_kernel(const float* __restrict__ blockSums, int nblocks,
                                float* __restrict__ out)
{
    if (threadIdx.x == 0 && blockIdx.x == 0) {
        float s = 0.0f;
        for (int i = 0; i < nblocks; ++i) s += blockSums[i];   // fixed order
        out[0] = s * (1.0f / (float)ROWS_TOTAL);
    }
}

extern "C" void kernel_launch(void* const* d_in, const int* in_sizes, int n_in,
                              void* d_out, int out_size, void* d_ws, size_t ws_size,
                              hipStream_t stream) {
    const float* cage    = (const float*)d_in[0];  // (4,128,3) f32
    const float* shape   = (const float*)d_in[1];  // (4,20000,3) f32
    const float* normals = (const float*)d_in[2];  // (4,20000,3) f32
    float* out = (float*)d_out;

    const size_t dpartBytes = (size_t)ROWS_TOTAL * NSPLIT * sizeof(float);  // 160 KB
    const size_t ipartBytes = (size_t)ROWS_TOTAL * NSPLIT * sizeof(int);    // 160 KB
    const int mergeBlocks   = ROWS_TOTAL / 256;                             // 20
    const size_t needA = dpartBytes + ipartBytes + mergeBlocks * sizeof(float);

    if (ws_size >= needA) {
        // Path A: 8-way N-split -> 320 blocks of WMMA scan, then merge.
        float* dpart     = (float*)d_ws;
        int*   ipart     = (int*)((char*)d_ws + dpartBytes);
        float* blockSums = (float*)((char*)d_ws + dpartBytes + ipartBytes);
        split_scan_kernel<<<BATCHES * BLOCKS_PER_BATCH * NSPLIT, 256, 0, stream>>>(
            cage, shape, dpart, ipart);
        merge_kernel<<<mergeBlocks, 256, 0, stream>>>(
            cage, shape, normals, dpart, ipart, blockSums);
        finalize_kernel<<<1, 32, 0, stream>>>(blockSums, mergeBlocks, out);
    } else {
        // Path B: single-pass fallback, needs only 40 floats of workspace.
        float* blockSums = (float*)d_ws;
        const int nb = BATCHES * BLOCKS_PER_BATCH;  // 40
        inside_loss_kernel<<<nb, 256, 0, stream>>>(cage, shape, normals, blockSums);
        finalize_kernel<<<1, 32, 0, stream>>>(blockSums, nb, out);
    }
}